// NLayer_78881369358594
// MI455X (gfx1250) — compile-verified
//
#include <hip/hip_runtime.h>
#include <hip/hip_bf16.h>
#include <math.h>

typedef __attribute__((ext_vector_type(2))) float v2f;
typedef __attribute__((ext_vector_type(8))) float v8f;

#define B_    2
#define V_    20000
#define C_    9
#define KS_   9
#define OUT_  16
#define NB_   16
#define KPAD  84          // 81 (= KS_*C_) padded to multiple of 4
#define SSTR  85          // odd LDS row stride -> conflict-free A fetches
#define WAVES 4
#define TPB   (WAVES * 32)
#define MTOT  (B_ * V_)

__global__ __launch_bounds__(TPB)
void nlayer_attn_wmma(const float* __restrict__ x,
                      const float* __restrict__ W,
                      const int*   __restrict__ adj,
                      float*       __restrict__ out)
{
    __shared__ float sWf[KPAD * OUT_];        // Wf[j][o], j = k*9 + c, zero-padded to 84
    __shared__ float sS [WAVES][32][SSTR];    // per-wave S rows (A matrix)
    __shared__ float sInv[WAVES][32];         // 1/deg per node

    const int tid  = threadIdx.x;
    const int wave = tid >> 5;
    const int lane = tid & 31;

    // ---- cooperative load of Wf into LDS: Wf[k*9+c][o] = W[c][k][o], pad rows = 0
    for (int idx = tid; idx < KPAD * OUT_; idx += TPB) {
        int j = idx >> 4;           // 0..83
        int o = idx & 15;
        float w = 0.0f;
        if (j < KS_ * C_) {
            int k = j / C_;
            int c = j - k * C_;
            w = W[(c * KS_ + k) * OUT_ + o];
        }
        sWf[idx] = w;
    }

    // ---- phase 1: one node per lane, accumulate S[k][c] = sum_n q[n,k] * xn[c]
    int m = blockIdx.x * TPB + tid;
    if (m >= MTOT) m = MTOT - 1;              // clamp; stores guarded below
    const int b = m / V_;
    const int v = m - b * V_;

    float xv[C_];
    const float* xrow = x + (size_t)m * C_;
#pragma unroll
    for (int c = 0; c < C_; ++c) xv[c] = xrow[c];

    float S[KS_][C_];
#pragma unroll
    for (int k = 0; k < KS_; ++k)
#pragma unroll
        for (int c = 0; c < C_; ++c) S[k][c] = 0.0f;

    int deg = 0;
    const int* arow = adj + (size_t)v * NB_;
#pragma unroll
    for (int n = 0; n < NB_; ++n) {
        int a = arow[n];
        if (a != 0) {
            ++deg;
            const float* xnrow = x + ((size_t)b * V_ + (size_t)(a - 1)) * C_;
            float xn[C_], d[C_];
            float mx = -INFINITY;
#pragma unroll
            for (int c = 0; c < C_; ++c) {
                xn[c] = xnrow[c];
                d[c]  = xv[c] - xn[c];
                mx    = fmaxf(mx, d[c]);
            }
            float e[C_];
            float s = 0.0f;
#pragma unroll
            for (int c = 0; c < C_; ++c) { e[c] = __expf(d[c] - mx); s += e[c]; }
            float inv = 1.0f / s;
#pragma unroll
            for (int k = 0; k < KS_; ++k) {
                float qk = e[k] * inv;   // C_ == KS_ : softmax axis doubles as k axis
#pragma unroll
                for (int c = 0; c < C_; ++c) S[k][c] = fmaf(qk, xn[c], S[k][c]);
            }
        }
    }

    // ---- spill S rows (+ zero pad) and 1/deg to LDS
#pragma unroll
    for (int k = 0; k < KS_; ++k)
#pragma unroll
        for (int c = 0; c < C_; ++c) sS[wave][lane][k * C_ + c] = S[k][c];
    sS[wave][lane][81] = 0.0f;
    sS[wave][lane][82] = 0.0f;
    sS[wave][lane][83] = 0.0f;
    sInv[wave][lane] = (deg > 0) ? (1.0f / (float)deg) : 0.0f;

    __syncthreads();

    // ---- phase 2: per wave, two 16x16 f32 WMMA tiles over K = 84
    const int half = lane >> 4;               // 0: K+0/K+1, 1: K+2/K+3
    const int lmod = lane & 15;               // M (A) / N (B,C/D) position

    v8f acc0 = {};
    v8f acc1 = {};
#pragma unroll
    for (int kb = 0; kb < KPAD; kb += 4) {
        const int kk = kb + 2 * half;
        v2f a0, a1, bb;
        a0.x = sS[wave][lmod][kk];        a0.y = sS[wave][lmod][kk + 1];
        a1.x = sS[wave][16 + lmod][kk];   a1.y = sS[wave][16 + lmod][kk + 1];
        bb.x = sWf[kk * OUT_ + lmod];     bb.y = sWf[(kk + 1) * OUT_ + lmod];
        acc0 = __builtin_amdgcn_wmma_f32_16x16x4_f32(
                   false, a0, false, bb, (short)0, acc0, false, false);
        acc1 = __builtin_amdgcn_wmma_f32_16x16x4_f32(
                   false, a1, false, bb, (short)0, acc1, false, false);
    }

    // ---- epilogue: scale by 1/deg, ReLU, guarded store
    const int waveBase = blockIdx.x * TPB + wave * 32;
#pragma unroll
    for (int i = 0; i < 8; ++i) {
        int r0 = i + 8 * half;                // tile-0 row within wave
        int m0 = waveBase + r0;
        if (m0 < MTOT)
            out[(size_t)m0 * OUT_ + lmod] = fmaxf(acc0[i] * sInv[wave][r0], 0.0f);
        int r1 = 16 + i + 8 * half;           // tile-1 row within wave
        int m1 = waveBase + r1;
        if (m1 < MTOT)
            out[(size_t)m1 * OUT_ + lmod] = fmaxf(acc1[i] * sInv[wave][r1], 0.0f);
    }
}

extern "C" void kernel_launch(void* const* d_in, const int* in_sizes, int n_in,
                              void* d_out, int out_size, void* d_ws, size_t ws_size,
                              hipStream_t stream)
{
    const float* x   = (const float*)d_in[0];   // (B, V, C)   f32
    const float* W   = (const float*)d_in[1];   // (C, KS, OUT) f32
    const int*   adj = (const int*)  d_in[2];   // (V, NB)     i32
    float*       out = (float*)d_out;           // (B, V, OUT) f32

    const int blocks = (MTOT + TPB - 1) / TPB;  // 313
    hipLaunchKernelGGL(nlayer_attn_wmma, dim3(blocks), dim3(TPB), 0, stream,
                       x, W, adj, out);
}